// UnifierMnrlLoss_80376017977624
// MI455X (gfx1250) — compile-verified
//
#include <hip/hip_runtime.h>
#include <math.h>

typedef __attribute__((ext_vector_type(2))) float v2f;
typedef __attribute__((ext_vector_type(4))) float v4f;
typedef __attribute__((ext_vector_type(8))) float v8f;

#define NB 2048      // query rows (B)
#define NK 4096      // doc rows (K)
#define ND 256       // full feature dim
#define NDIMS 4      // matryoshka levels: 64,128,192,256
#define CONFLICT_TOL 2.0e-5f   // atol + rtol*|1.0| from jnp.isclose

#define WMMA_F32(A, B, C) \
  __builtin_amdgcn_wmma_f32_16x16x4_f32(false, (A), false, (B), (short)0, (C), false, false)

__device__ __forceinline__ float redSum32(float v) {
  v += __shfl_xor(v, 1, 32);
  v += __shfl_xor(v, 2, 32);
  v += __shfl_xor(v, 4, 32);
  v += __shfl_xor(v, 8, 32);
  v += __shfl_xor(v, 16, 32);
  return v;
}
__device__ __forceinline__ float redSum16(float v) {
  v += __shfl_xor(v, 1, 32);
  v += __shfl_xor(v, 2, 32);
  v += __shfl_xor(v, 4, 32);
  v += __shfl_xor(v, 8, 32);
  return v;
}

// -------- Kernel 1: per-row prefix norms + q·d diagonal prefix dots ----------
__global__ void prep_norms(const float* __restrict__ q, const float* __restrict__ d,
                           float* __restrict__ inv_nq, float* __restrict__ inv_nd,
                           float* __restrict__ qd_raw)
{
  const int gwave = (int)((blockIdx.x * blockDim.x + threadIdx.x) >> 5);
  const int lane  = threadIdx.x & 31;
  if (gwave < NK) {
    const float* row = d + (size_t)gwave * ND;
    float cs[NDIMS] = {0.f, 0.f, 0.f, 0.f};
#pragma unroll
    for (int t = 0; t < 8; ++t) { float x = row[lane + 32 * t]; cs[t >> 1] += x * x; }
    float run = 0.f;
#pragma unroll
    for (int c = 0; c < NDIMS; ++c) {
      run += redSum32(cs[c]);
      if (lane == 0) inv_nd[c * NK + gwave] = 1.0f / fmaxf(sqrtf(run), 1e-12f);
    }
  } else if (gwave < NK + NB) {
    const int i = gwave - NK;
    const float* rq = q + (size_t)i * ND;
    const float* rd = d + (size_t)i * ND;
    float cs[NDIMS] = {0.f, 0.f, 0.f, 0.f};
    float cd[NDIMS] = {0.f, 0.f, 0.f, 0.f};
#pragma unroll
    for (int t = 0; t < 8; ++t) {
      float a = rq[lane + 32 * t], b = rd[lane + 32 * t];
      cs[t >> 1] += a * a;
      cd[t >> 1] += a * b;
    }
    float run = 0.f, rund = 0.f;
#pragma unroll
    for (int c = 0; c < NDIMS; ++c) {
      run  += redSum32(cs[c]);
      rund += redSum32(cd[c]);
      if (lane == 0) {
        inv_nq[c * NB + i] = 1.0f / fmaxf(sqrtf(run), 1e-12f);
        qd_raw[c * NB + i] = rund;
      }
    }
  }
}

// -------- Kernel 2: thresholds thr = diag(qk) + 0.1, zero the accumulators ---
__global__ void prep_thr(const float* __restrict__ inv_nd, const float* __restrict__ inv_nq,
                         const float* __restrict__ qd_raw,
                         float* __restrict__ thrArr, float* __restrict__ sum_exp)
{
  const int idx = blockIdx.x * blockDim.x + threadIdx.x;
  if (idx < NDIMS * NB) {
    const int c = idx / NB, i = idx - c * NB;
    thrArr[idx]  = qd_raw[idx] * inv_nq[idx] * inv_nd[c * NK + i] + 0.1f;
    sum_exp[idx] = 0.0f;
  }
}

// -------- Checkpoint: cumulative raw dots -> masked exp contributions --------
// Processes BOTH of a wave's 16x16 subtiles so per-row constants load once.
// C/D layout: VGPR e -> row (e or e+8 for upper half-lanes), col = lane&15.
template <int MODE>
__device__ __forceinline__ void checkpoint_pair(
    int c, int iBase, int jBase2, int lane,
    const v8f& accS0, const v8f& accD0, const v8f& accS1, const v8f& accD1,
    const float* __restrict__ inv_nq, const float* __restrict__ inv_nd,
    const float* __restrict__ thrArr, bool colsKQ,
    float* rowAccC, float* colAccC, int mLocalBase, int nLocalBase)
{
  const int half = lane >> 4, m16 = lane & 15;
  const int j0 = jBase2 + m16, j1 = j0 + 16;
  const float nj0 = (MODE == 0) ? inv_nd[c * NK + j0] : inv_nq[c * NB + j0];
  const float nj1 = (MODE == 0) ? inv_nd[c * NK + j1] : inv_nq[c * NB + j1];
  const float tj0 = colsKQ ? thrArr[c * NB + j0] : 0.0f;
  const float tj1 = colsKQ ? thrArr[c * NB + j1] : 0.0f;
  float colC0 = 0.0f, colC1 = 0.0f;
#pragma unroll
  for (int e = 0; e < 8; ++e) {
    const int i = iBase + e + 8 * half;
    const float inq_i = inv_nq[c * NB + i];   // uniform across 16-lane group
    const float thr_i = thrArr[c * NB + i];
    float rowC;
    if (MODE == 0) {
      const float ind_i = inv_nd[c * NK + i];            // i < 2048 (pos rows)
      const float s0 = accS0[e] * inq_i * nj0;           // qk / kq values
      const float s1 = accS1[e] * inq_i * nj1;
      const float m0 = accD0[e] * ind_i * nj0;           // sim / kk values
      const float m1 = accD1[e] * ind_i * nj1;
      const bool  dg0 = (i == j0), dg1 = (i == j1);
      const bool  cf0 = (!dg0) && (fabsf(m0 - 1.0f) <= CONFLICT_TOL);
      const bool  cf1 = (!dg1) && (fabsf(m1 - 1.0f) <= CONFLICT_TOL);
      rowC  = ((!cf0) && (s0 <= thr_i)) ? __expf(s0 - thr_i) : 0.0f;     // qk
      rowC += ((!cf1) && (s1 <= thr_i)) ? __expf(s1 - thr_i) : 0.0f;
      if (colsKQ) {
        rowC  += ((!dg0) && (m0 <= thr_i)) ? __expf(m0 - thr_i) : 0.0f;  // kk
        rowC  += ((!dg1) && (m1 <= thr_i)) ? __expf(m1 - thr_i) : 0.0f;
        colC0 += ((!dg0) && (s0 <= tj0)) ? __expf(s0 - tj0) : 0.0f;      // kq
        colC1 += ((!dg1) && (s1 <= tj1)) ? __expf(s1 - tj1) : 0.0f;
      }
    } else {
      const float s0 = accS0[e] * inq_i * nj0;           // qq values
      const float s1 = accS1[e] * inq_i * nj1;
      rowC  = ((i != j0) && (s0 <= thr_i)) ? __expf(s0 - thr_i) : 0.0f;
      rowC += ((i != j1) && (s1 <= thr_i)) ? __expf(s1 - thr_i) : 0.0f;
    }
    rowC = redSum16(rowC);   // sum the 16 columns per half-lane group
    if (m16 == 0) atomicAdd(&rowAccC[mLocalBase + e + 8 * half], rowC);
  }
  if (colsKQ) {
    colC0 += __shfl_xor(colC0, 16, 32);   // halves hold the same column set
    colC1 += __shfl_xor(colC1, 16, 32);
    if (half == 0) {
      atomicAdd(&colAccC[nLocalBase + m16], colC0);
      atomicAdd(&colAccC[nLocalBase + 16 + m16], colC1);
    }
  }
}

// -------- Kernels 3/4: fused WMMA GEMM + streaming masked exp accumulation ---
// MODE 0: rows=q & d (QK + DD share B fragments), cols=d (64x64 tile/block)
// MODE 1: rows=q, cols=q (QQ)
// K runs in 4 chunks of 64; accumulators are cumulative so every matryoshka
// prefix dim is a free snapshot.
// K-slice repartition for wide loads: half 0 loads cols [8t,8t+3] as b128,
// half 1 loads [8t+4,8t+7]; slice LO contracts {8t,8t+1,8t+4,8t+5}, slice HI
// {8t+2,8t+3,8t+6,8t+7} — A and B use the same assignment, so the GEMM sum is
// unchanged while every load is a full 16-byte b128.
template <int MODE>
__global__ void __launch_bounds__(256)
fused_scores(const float* __restrict__ q, const float* __restrict__ d,
             const float* __restrict__ inv_nq, const float* __restrict__ inv_nd,
             const float* __restrict__ thrArr, float* __restrict__ sum_exp)
{
  __shared__ float rowAcc[NDIMS][64];
  __shared__ float colAcc[NDIMS][64];
  const int tid = threadIdx.x;
  for (int t = tid; t < NDIMS * 64; t += 256) {
    (&rowAcc[0][0])[t] = 0.0f;
    (&colAcc[0][0])[t] = 0.0f;
  }
  __syncthreads();

  const int lane = tid & 31, wave = tid >> 5;
  const int half = lane >> 4, m16 = lane & 15;
  const int m_sub = wave & 3, n_pair = wave >> 2;
  const int iBase = blockIdx.y * 64 + m_sub * 16;
  const int jBase = blockIdx.x * 64 + n_pair * 32;
  const bool colsKQ = (MODE == 0) && ((int)blockIdx.x * 64 < NB);

  const float* __restrict__ colMat = (MODE == 0) ? d : q;
  const float* aQrow = q + (size_t)(iBase + m16) * ND;
  const float* aDrow = d + (size_t)(iBase + m16) * ND;
  const float* bRow0 = colMat + (size_t)(jBase + m16) * ND;
  const float* bRow1 = colMat + (size_t)(jBase + 16 + m16) * ND;

  v8f accS0 = {}, accS1 = {}, accD0 = {}, accD1 = {};

  for (int c = 0; c < NDIMS; ++c) {
    const int kc = c * 64 + 4 * half;
#pragma unroll 2
    for (int t = 0; t < 8; ++t) {
      const int k0 = kc + 8 * t;
      const v4f aq = *(const v4f*)(aQrow + k0);
      const v4f b0 = *(const v4f*)(bRow0 + k0);
      const v4f b1 = *(const v4f*)(bRow1 + k0);
      const v2f aqLo = {aq.x, aq.y}, aqHi = {aq.z, aq.w};
      const v2f b0Lo = {b0.x, b0.y}, b0Hi = {b0.z, b0.w};
      const v2f b1Lo = {b1.x, b1.y}, b1Hi = {b1.z, b1.w};
      accS0 = WMMA_F32(aqLo, b0Lo, accS0);
      accS0 = WMMA_F32(aqHi, b0Hi, accS0);
      accS1 = WMMA_F32(aqLo, b1Lo, accS1);
      accS1 = WMMA_F32(aqHi, b1Hi, accS1);
      if (MODE == 0) {
        const v4f ad = *(const v4f*)(aDrow + k0);
        const v2f adLo = {ad.x, ad.y}, adHi = {ad.z, ad.w};
        accD0 = WMMA_F32(adLo, b0Lo, accD0);
        accD0 = WMMA_F32(adHi, b0Hi, accD0);
        accD1 = WMMA_F32(adLo, b1Lo, accD1);
        accD1 = WMMA_F32(adHi, b1Hi, accD1);
      }
    }
    // Snapshot at prefix dim 64*(c+1).
    checkpoint_pair<MODE>(c, iBase, jBase, lane, accS0, accD0, accS1, accD1,
                          inv_nq, inv_nd, thrArr, colsKQ,
                          &rowAcc[c][0], &colAcc[c][0], m_sub * 16, n_pair * 32);
  }
  __syncthreads();
  for (int t = tid; t < NDIMS * 64; t += 256) {
    const int c = t >> 6, r = t & 63;
    atomicAdd(&sum_exp[c * NB + blockIdx.y * 64 + r], rowAcc[c][r]);
    if (colsKQ) atomicAdd(&sum_exp[c * NB + blockIdx.x * 64 + r], colAcc[c][r]);
  }
}

// -------- Kernel 5: loss = mean over dims,rows of (0.1 + log(sum_exp)) -------
__global__ void finalize(const float* __restrict__ sum_exp, float* __restrict__ out)
{
  __shared__ float red[256];
  float acc = 0.0f;
  for (int t = threadIdx.x; t < NDIMS * NB; t += 256)
    acc += 0.1f + __logf(sum_exp[t]);
  red[threadIdx.x] = acc;
  __syncthreads();
  for (int s = 128; s > 0; s >>= 1) {
    if ((int)threadIdx.x < s) red[threadIdx.x] += red[threadIdx.x + s];
    __syncthreads();
  }
  if (threadIdx.x == 0) out[0] = red[0] / (float)(NDIMS * NB);
}

extern "C" void kernel_launch(void* const* d_in, const int* in_sizes, int n_in,
                              void* d_out, int out_size, void* d_ws, size_t ws_size,
                              hipStream_t stream)
{
  (void)in_sizes; (void)n_in; (void)out_size; (void)ws_size;
  const float* q = (const float*)d_in[0];   // reps_q: 2048x256 f32
  const float* d = (const float*)d_in[1];   // reps_d: 4096x256 f32

  // Workspace layout (floats): total 49152 floats = 192 KB
  float* ws      = (float*)d_ws;
  float* inv_nd  = ws;                       // [4][4096]
  float* inv_nq  = inv_nd + NDIMS * NK;      // [4][2048]
  float* qd_raw  = inv_nq + NDIMS * NB;      // [4][2048]
  float* thrArr  = qd_raw + NDIMS * NB;      // [4][2048]
  float* sum_exp = thrArr + NDIMS * NB;      // [4][2048]

  const int waves = NK + NB;                 // one wave per row
  prep_norms<<<(waves * 32 + 255) / 256, 256, 0, stream>>>(q, d, inv_nq, inv_nd, qd_raw);
  prep_thr<<<(NDIMS * NB + 255) / 256, 256, 0, stream>>>(inv_nd, inv_nq, qd_raw, thrArr, sum_exp);

  dim3 g0(NK / 64, NB / 64);                 // QK + DD over 2048x4096
  fused_scores<0><<<g0, 256, 0, stream>>>(q, d, inv_nq, inv_nd, thrArr, sum_exp);
  dim3 g1(NB / 64, NB / 64);                 // QQ over 2048x2048
  fused_scores<1><<<g1, 256, 0, stream>>>(q, d, inv_nq, inv_nd, thrArr, sum_exp);

  finalize<<<1, 256, 0, stream>>>(sum_exp, (float*)d_out);
}